// DMCR_57088705298541
// MI455X (gfx1250) — compile-verified
//
#include <hip/hip_runtime.h>

typedef float v2f __attribute__((ext_vector_type(2)));
typedef float v8f __attribute__((ext_vector_type(8)));

#define U_CNT 40000
#define I_CNT 40000
#define N_TOT 80000
#define C_CNT 5
#define D_DIM 128
#define A_DIM 64
#define L_CNT 2
#define E_CNT 1280000
#define CD (C_CNT * D_DIM)          /* 640 */
#define NEG_SLOPE 0.01f

// ---------------------------------------------------------------------------
// init: ego = concat(user, item); bank = ego     (float4 vectorized)
// ---------------------------------------------------------------------------
__global__ __launch_bounds__(256) void dmcr_init_kernel(
    const float4* __restrict__ user, const float4* __restrict__ item,
    float4* __restrict__ ego, float4* __restrict__ bank, int nUser4, int nTot4)
{
    int t = blockIdx.x * 256 + threadIdx.x;
    if (t >= nTot4) return;
    float4 v = (t < nUser4) ? user[t] : item[t - nUser4];
    ego[t]  = v;
    bank[t] = v;
}

// ---------------------------------------------------------------------------
// scatter: one wave32 per edge.  feats[row, c, :] += val * ego[col, c, :]
// 128 floats per edge = float4 per lane; hardware f32 atomics.
// Per-criterion working set (2 x 41 MB) is L2-resident (192 MB).
// ---------------------------------------------------------------------------
__global__ __launch_bounds__(256) void dmcr_scatter_kernel(
    const float* __restrict__ ego, const float* __restrict__ vals,
    const int* __restrict__ rows, const int* __restrict__ cols,
    float* __restrict__ feats)
{
    int wid = blockIdx.x * 8 + (threadIdx.x >> 5);
    if (wid >= C_CNT * E_CNT) return;
    int lane = threadIdx.x & 31;
    int c    = wid / E_CNT;                 // vals/rows/cols are (C,E) flat
    float v  = vals[wid];
    int col  = cols[wid];
    int row  = rows[wid];
    const float4* src = (const float4*)(ego + (size_t)col * CD + c * D_DIM);
    float4 x = src[lane];
    float* dst = feats + (size_t)row * CD + c * D_DIM + (lane << 2);
    unsafeAtomicAdd(dst + 0, v * x.x);
    unsafeAtomicAdd(dst + 1, v * x.y);
    unsafeAtomicAdd(dst + 2, v * x.z);
    unsafeAtomicAdd(dst + 3, v * x.w);
}

// ---------------------------------------------------------------------------
// WMMA fp32 GEMM:  Y = epilogue( X @ W )
//   X : (M = N*C, 128) row-major, W : (128,128) row-major
//   block = 256 thr = 8 waves; block tile = 64 rows x 128 cols.
//   Each wave owns a 16-col panel: B fragments (128x16) preloaded to 64 VGPRs
//   once, then reused across 4 sub-tiles of 16 rows -> 128 WMMAs per wave.
//   MODE 0: Y *= crit[row%C, col]        (combined = gf * crit_vec)
//   MODE 1: Y = leaky_relu(Y, 0.01)
// ---------------------------------------------------------------------------
template <int MODE>
__global__ __launch_bounds__(256) void dmcr_gemm_wmma_kernel(
    const float* __restrict__ X, const float* __restrict__ W,
    float* __restrict__ Y, const float* __restrict__ crit)
{
    __shared__ float As[64 * 128];          // 32 KB A tile (contiguous rows)
    int tid = threadIdx.x;
    size_t rowBase = (size_t)blockIdx.x * 64;

    // cooperative A-tile load: 8192 floats = 2048 float4, 8 per thread
    {
        const float4* Xt4 = (const float4*)(X + rowBase * 128);
        float4* As4 = (float4*)As;
        #pragma unroll
        for (int i = 0; i < 8; ++i) As4[tid + i * 256] = Xt4[tid + i * 256];
    }
    __syncthreads();

    int wave = tid >> 5, lane = tid & 31;
    int colBase = wave * 16;
    int m    = lane & 15;                   // A: row within tile / B,C,D: col
    int kSel = (lane >> 4) * 2;             // lanes 16-31 carry K+2,K+3

    // preload B panel (128 x 16) into registers: 32 v2f fragments
    v2f bf[32];
    #pragma unroll
    for (int k = 0; k < 32; ++k) {
        int ka = k * 4 + kSel;
        bf[k].x = W[ka * 128 + colBase + m];
        bf[k].y = W[(ka + 1) * 128 + colBase + m];
    }

    int cb = MODE == 0 ? ((int)rowBase % C_CNT) : 0;   // uniform scalar

    #pragma unroll
    for (int s = 0; s < 4; ++s) {
        const float* Arow = As + (s * 16 + m) * 128;
        v8f acc = {0.f,0.f,0.f,0.f,0.f,0.f,0.f,0.f};
        #pragma unroll
        for (int k = 0; k < 32; ++k) {
            int ka = k * 4 + kSel;
            v2f a;
            a.x = Arow[ka];
            a.y = Arow[ka + 1];
            acc = __builtin_amdgcn_wmma_f32_16x16x4_f32(
                      false, a, false, bf[k], (short)0, acc, false, false);
        }
        #pragma unroll
        for (int r = 0; r < 8; ++r) {
            int mo   = r + ((lane >> 4) << 3);   // C/D: vgpr r -> M=r (+8 hi)
            int rloc = s * 16 + mo;
            int col  = colBase + m;
            float v  = acc[r];
            if (MODE == 0) {
                int c = (cb + rloc) % C_CNT;     // 32-bit, value < 85
                v *= crit[c * D_DIM + col];
            } else {
                v = v > 0.f ? v : NEG_SLOPE * v;
            }
            Y[(rowBase + rloc) * 128 + col] = v;
        }
    }
}

// ---------------------------------------------------------------------------
// attention: per block = 8 nodes, stacked rows staged in LDS.
//   raw[k,c] = sum_a tanh(S[c,:]·p1[k,:,a]) * p2[k,a]   (wave per job, lanes
//   split over a), softmax over c, ego[n,k,:] = sum_c w * S[c,:], bank += ego
// ---------------------------------------------------------------------------
#define NB 8
__global__ __launch_bounds__(256) void dmcr_attn_kernel(
    const float* __restrict__ stacked, const float* __restrict__ p1,
    const float* __restrict__ p2, float* __restrict__ ego,
    float* __restrict__ bank)
{
    __shared__ float S[NB * CD];            // 20 KB
    __shared__ float raw_s[NB * 25];
    __shared__ float wsm[NB * 25];
    int tid = threadIdx.x;
    size_t nb = (size_t)blockIdx.x * NB;

    for (int i = tid; i < NB * CD; i += 256)
        S[i] = stacked[nb * CD + i];
    __syncthreads();

    int wave = tid >> 5, lane = tid & 31;
    int a0 = lane, a1 = lane + 32;
    for (int j = wave; j < NB * 25; j += 8) {
        int n = j / 25, pr = j % 25;
        int k = pr / 5, c = pr % 5;
        const float* Sc = S + n * CD + c * D_DIM;
        const float* P  = p1 + (size_t)k * D_DIM * A_DIM;
        float acc0 = 0.f, acc1 = 0.f;
        #pragma unroll 8
        for (int d = 0; d < D_DIM; ++d) {
            float sv = Sc[d];
            acc0 += sv * P[d * A_DIM + a0];
            acc1 += sv * P[d * A_DIM + a1];
        }
        float r = tanhf(acc0) * p2[k * A_DIM + a0]
                + tanhf(acc1) * p2[k * A_DIM + a1];
        #pragma unroll
        for (int off = 16; off > 0; off >>= 1)
            r += __shfl_xor(r, off, 32);
        if (lane == 0) raw_s[j] = r;
    }
    __syncthreads();

    if (tid < NB * 5) {                     // softmax over c for each (n,k)
        int n = tid / 5, k = tid % 5;
        const float* rw = raw_s + n * 25 + k * 5;
        float mx = rw[0];
        #pragma unroll
        for (int c = 1; c < 5; ++c) mx = fmaxf(mx, rw[c]);
        float e[5], s = 0.f;
        #pragma unroll
        for (int c = 0; c < 5; ++c) { e[c] = __expf(rw[c] - mx); s += e[c]; }
        float inv = 1.f / s;
        #pragma unroll
        for (int c = 0; c < 5; ++c) wsm[n * 25 + k * 5 + c] = e[c] * inv;
    }
    __syncthreads();

    for (int i = tid; i < NB * CD; i += 256) {
        int n = i / CD, r = i % CD;
        int k = r >> 7, d = r & 127;
        const float* w = wsm + n * 25 + k * 5;
        const float* Sn = S + n * CD;
        float acc = 0.f;
        #pragma unroll
        for (int c = 0; c < 5; ++c) acc += w[c] * Sn[c * D_DIM + d];
        size_t g = nb * CD + i;
        ego[g] = acc;
        bank[g] += acc;
    }
}

// ---------------------------------------------------------------------------
// crit update: cout = cin @ W   (5x128 @ 128x128), single launch of 640 dots
// ---------------------------------------------------------------------------
__global__ __launch_bounds__(256) void dmcr_critmm_kernel(
    const float* __restrict__ cin, const float* __restrict__ W,
    float* __restrict__ cout)
{
    int t = blockIdx.x * 256 + threadIdx.x;
    if (t >= CD) return;
    int c = t >> 7, d = t & 127;
    float acc = 0.f;
    #pragma unroll 8
    for (int e = 0; e < D_DIM; ++e) acc += cin[c * D_DIM + e] * W[e * D_DIM + d];
    cout[t] = acc;
}

// ---------------------------------------------------------------------------
// final: out = [bank[:U]/3 | bank[U:]/3 | zeros(640) | mean(crit_hist)]
// ---------------------------------------------------------------------------
__global__ __launch_bounds__(256) void dmcr_final_kernel(
    const float* __restrict__ bank, const float* __restrict__ c0,
    const float* __restrict__ c1, const float* __restrict__ c2,
    float* __restrict__ out, size_t nBank, size_t total)
{
    size_t t = (size_t)blockIdx.x * 256 + threadIdx.x;
    if (t >= total) return;
    const float third = 1.f / 3.f;
    if (t < nBank) {
        out[t] = bank[t] * third;
    } else if (t < nBank + CD) {
        out[t] = 0.f;                       // padded zero row of item_out
    } else {
        size_t i = t - nBank - CD;
        out[t] = (c0[i] + c1[i] + c2[i]) * third;
    }
}

// ---------------------------------------------------------------------------
extern "C" void kernel_launch(void* const* d_in, const int* in_sizes, int n_in,
                              void* d_out, int out_size, void* d_ws, size_t ws_size,
                              hipStream_t stream) {
    const float* user   = (const float*)d_in[0];
    const float* item   = (const float*)d_in[1];
    const float* critE  = (const float*)d_in[2];
    const float* gcnW   = (const float*)d_in[3];
    const float* gcnL   = (const float*)d_in[4];   // (2,128,128)
    const float* critL  = (const float*)d_in[5];   // (2,128,128)
    const float* p1     = (const float*)d_in[6];   // (5,128,64)
    const float* p2     = (const float*)d_in[7];   // (5,64)
    const float* vals   = (const float*)d_in[8];   // (5,E)
    const int*   rows   = (const int*)d_in[9];
    const int*   cols   = (const int*)d_in[10];
    float* out = (float*)d_out;

    const size_t NCD = (size_t)N_TOT * CD;         // 51,200,000 floats
    float* ego   = (float*)d_ws;
    float* feats = ego + NCD;
    float* bank  = feats + NCD;
    float* crit[3];
    crit[0] = bank + NCD;
    crit[1] = crit[0] + CD;
    crit[2] = crit[1] + CD;

    // init ego/bank and crit_hist[0]
    {
        int nUser4 = (U_CNT * CD) / 4;
        int nTot4  = (int)(NCD / 4);
        dmcr_init_kernel<<<nTot4 / 256, 256, 0, stream>>>(
            (const float4*)user, (const float4*)item,
            (float4*)ego, (float4*)bank, nUser4, nTot4);
        hipMemcpyAsync(crit[0], critE, CD * sizeof(float),
                       hipMemcpyDeviceToDevice, stream);
    }

    const int scatterBlocks = (C_CNT * E_CNT + 7) / 8;       // 800,000
    const int gemmBlocks    = (N_TOT * C_CNT) / 64;          // 6,250
    const int attnBlocks    = N_TOT / NB;                    // 10,000

    for (int l = 0; l < L_CNT; ++l) {
        hipMemsetAsync(feats, 0, NCD * sizeof(float), stream);
        dmcr_scatter_kernel<<<scatterBlocks, 256, 0, stream>>>(
            ego, vals, rows, cols, feats);
        // gf = feats @ gcnW ; combined = gf * crit_vec  -> ego
        dmcr_gemm_wmma_kernel<0><<<gemmBlocks, 256, 0, stream>>>(
            feats, gcnW, ego, crit[l]);
        // stacked = leaky_relu(combined @ gcn_layers[l]) -> feats
        dmcr_gemm_wmma_kernel<1><<<gemmBlocks, 256, 0, stream>>>(
            ego, gcnL + (size_t)l * D_DIM * D_DIM, feats, nullptr);
        // attention: feats(stacked) -> ego(new), bank += ego
        dmcr_attn_kernel<<<attnBlocks, 256, 0, stream>>>(
            feats, p1, p2, ego, bank);
        // crit_hist[l+1] = crit_hist[l] @ crit_layers[l]
        dmcr_critmm_kernel<<<3, 256, 0, stream>>>(
            crit[l], critL + (size_t)l * D_DIM * D_DIM, crit[l + 1]);
    }

    const size_t total = NCD + 2 * CD;             // 51,201,280 == out_size
    dmcr_final_kernel<<<(unsigned)((total + 255) / 256), 256, 0, stream>>>(
        bank, crit[0], crit[1], crit[2], out, NCD, total);
}